// GCN_43568148250684
// MI455X (gfx1250) — compile-verified
//
#include <hip/hip_runtime.h>
#include <math.h>

typedef __attribute__((ext_vector_type(2))) float v2f;
typedef __attribute__((ext_vector_type(8))) float v8f;

#define N_NODES 100000
#define N_EDGES 3200000
#define NFEAT   512
#define NHID    256
#define NCLS    7
#define NOUT2   9   // 7 (W2 head) + 2 (W3 head), fused
#define MT      5   // M-tiles (16 rows each) per wave in GEMM1
#define NT      2   // N-tiles (16 cols each) per wave in GEMM1

// ---------------------------------------------------------------------------
// CSR construction (so SpMM needs no float atomics)
// ---------------------------------------------------------------------------
__global__ void k_zero_counts(int* __restrict__ counts) {
    int i = blockIdx.x * blockDim.x + threadIdx.x;
    if (i < N_NODES) counts[i] = 0;
}

__global__ void k_hist(const int* __restrict__ erow, int* __restrict__ counts) {
    int e = blockIdx.x * blockDim.x + threadIdx.x;
    if (e < N_EDGES) atomicAdd(&counts[erow[e]], 1);
}

// single-workgroup chunked Hillis-Steele scan over 100k counts -> exclusive row_start
__global__ void k_scan(const int* __restrict__ counts, int* __restrict__ row_start) {
    __shared__ int s[1024];
    __shared__ int s_run;
    int tid = threadIdx.x;
    if (tid == 0) s_run = 0;
    __syncthreads();
    for (int base = 0; base < N_NODES; base += 1024) {
        int v = (base + tid < N_NODES) ? counts[base + tid] : 0;
        s[tid] = v;
        __syncthreads();
        for (int off = 1; off < 1024; off <<= 1) {
            int t = (tid >= off) ? s[tid - off] : 0;
            __syncthreads();
            s[tid] += t;
            __syncthreads();
        }
        int run = s_run;
        if (base + tid < N_NODES) row_start[base + tid] = run + s[tid] - v;
        __syncthreads();
        if (tid == 0) s_run = run + s[1023];
        __syncthreads();
    }
    if (threadIdx.x == 0) row_start[N_NODES] = s_run;
}

__global__ void k_init_cursor(const int* __restrict__ row_start, int* __restrict__ cursor) {
    int i = blockIdx.x * blockDim.x + threadIdx.x;
    if (i < N_NODES) cursor[i] = row_start[i];
}

__global__ void k_scatter(const int* __restrict__ erow, int* __restrict__ cursor,
                          int* __restrict__ esort) {
    int e = blockIdx.x * blockDim.x + threadIdx.x;
    if (e < N_EDGES) {
        int pos = atomicAdd(&cursor[erow[e]], 1);
        esort[pos] = e;
    }
}

// ---------------------------------------------------------------------------
// GEMM1: h0 = x(100000x512) @ W1(512x256), full f32 via V_WMMA_F32_16X16X4_F32.
// Register-tiled: each wave computes a (5x16) x (2x16) = 80x32 output panel,
// so each 4-deep K step issues 5 A-loads(b64) + 4 B-loads(b32) for 10 WMMAs.
// Block = 8 waves covering all 256 N columns; grid.x = 100000/80 = 1250.
// ---------------------------------------------------------------------------
__global__ void k_gemm1_wmma(const float* __restrict__ A, const float* __restrict__ B,
                             float* __restrict__ C) {
    int tid  = threadIdx.x;
    int wave = tid >> 5;
    int lane = tid & 31;
    int l16  = lane & 15;
    int g    = lane >> 4;                      // half-wave group (0/1)
    int mbase = blockIdx.x * (MT * 16);        // first row of this block's M panel
    int nbase = wave * (NT * 16);              // first col of this wave's N panel

    const float* Ap = A + (size_t)(mbase + l16) * NFEAT + 2 * g;
    const float* Bp = B + (size_t)(2 * g) * NHID + nbase + l16;

    v8f zero = {};
    v8f acc[MT][NT];
#pragma unroll
    for (int i = 0; i < MT; ++i)
#pragma unroll
        for (int j = 0; j < NT; ++j) acc[i][j] = zero;

    for (int kk = 0; kk < NFEAT; kk += 4) {
        v2f a[MT], b[NT];
#pragma unroll
        for (int i = 0; i < MT; ++i) {
            // A 16x4 fragment: vgpr r holds A[m][k0+r] (contiguous pair -> b64)
            float2 t = *(const float2*)(Ap + (size_t)i * 16 * NFEAT);
            a[i].x = t.x; a[i].y = t.y;
        }
#pragma unroll
        for (int j = 0; j < NT; ++j) {
            // B 4x16 fragment: vgpr r holds B[k0+r][n]
            b[j].x = Bp[j * 16];
            b[j].y = Bp[j * 16 + NHID];
        }
#pragma unroll
        for (int i = 0; i < MT; ++i)
#pragma unroll
            for (int j = 0; j < NT; ++j)
                acc[i][j] = __builtin_amdgcn_wmma_f32_16x16x4_f32(
                    false, a[i], false, b[j], (short)0, acc[i][j], false, false);
        Ap += 4;
        Bp += 4 * NHID;
    }

    // D layout: vgpr r, half-wave g -> row = tile_row + r + 8*g, col = n
#pragma unroll
    for (int i = 0; i < MT; ++i) {
        int mrow = mbase + i * 16 + 8 * g;
#pragma unroll
        for (int j = 0; j < NT; ++j) {
            int n = nbase + j * 16 + l16;
#pragma unroll
            for (int r = 0; r < 8; ++r)
                C[(size_t)(mrow + r) * NHID + n] = acc[i][j][r];
        }
    }
}

// ---------------------------------------------------------------------------
// Big SpMM (CSR): one block per output row, 256 threads = 256 features.
// Edge metadata staged through LDS; every h0 gather is a coalesced 1KB row
// that stays L2-resident (h0 = 102.4 MB < 192 MB L2). Fused bias + relu.
// ---------------------------------------------------------------------------
__global__ void k_spmm_big(const int* __restrict__ row_start, const int* __restrict__ esort,
                           const int* __restrict__ ecol, const float* __restrict__ ew,
                           const float* __restrict__ h0, const float* __restrict__ b1,
                           float* __restrict__ h) {
    __shared__ int   s_col[64];
    __shared__ float s_w[64];
    int r   = blockIdx.x;
    int tid = threadIdx.x;
    int beg = row_start[r], end = row_start[r + 1];
    float acc = 0.f;
    for (int base = beg; base < end; base += 64) {
        int cnt = end - base; if (cnt > 64) cnt = 64;
        if (tid < cnt) {
            int e = esort[base + tid];
            s_col[tid] = ecol[e];
            s_w[tid]   = ew[e];
        }
        __syncthreads();
        for (int i = 0; i < cnt; ++i)
            acc += s_w[i] * h0[(size_t)s_col[i] * NHID + tid];
        __syncthreads();
    }
    float v = acc + b1[tid];
    h[(size_t)r * NHID + tid] = v > 0.f ? v : 0.f;
}

// ---------------------------------------------------------------------------
// Fused tiny GEMM: t12 = h(100000x256) @ [W2|W3](256x9). One wave per row,
// W2|W3 interleaved in LDS, lanes stride K, shuffle reduction.
// ---------------------------------------------------------------------------
__global__ void k_gemm_small(const float* __restrict__ h, const float* __restrict__ W2,
                             const float* __restrict__ W3, float* __restrict__ t12) {
    __shared__ float sW[NHID * NOUT2];
    for (int i = threadIdx.x; i < NHID * NCLS; i += blockDim.x)
        sW[(i / NCLS) * NOUT2 + (i % NCLS)] = W2[i];
    for (int i = threadIdx.x; i < NHID * 2; i += blockDim.x)
        sW[(i / 2) * NOUT2 + NCLS + (i % 2)] = W3[i];
    __syncthreads();
    int lane = threadIdx.x & 31;
    int r = blockIdx.x * 8 + (threadIdx.x >> 5);
    const float* hr = h + (size_t)r * NHID;
    float p[NOUT2] = {};
    for (int k = lane; k < NHID; k += 32) {
        float hv = hr[k];
        const float* wr = sW + k * NOUT2;
#pragma unroll
        for (int j = 0; j < NOUT2; ++j) p[j] += hv * wr[j];
    }
#pragma unroll
    for (int j = 0; j < NOUT2; ++j)
        for (int o = 16; o > 0; o >>= 1) p[j] += __shfl_xor(p[j], o, 32);
    if (lane == 0) {
#pragma unroll
        for (int j = 0; j < NOUT2; ++j) t12[(size_t)r * NOUT2 + j] = p[j];
    }
}

// ---------------------------------------------------------------------------
// Small SpMM (CSR) fused with biases + log_softmax(7) + log_softmax(2) +
// softmax(x1)[:, -1]. One wave per row; lanes parallel over edges.
// ---------------------------------------------------------------------------
__global__ void k_spmm_small_softmax(const int* __restrict__ row_start,
                                     const int* __restrict__ esort,
                                     const int* __restrict__ ecol,
                                     const float* __restrict__ ew,
                                     const float* __restrict__ t12,
                                     const float* __restrict__ b2,
                                     const float* __restrict__ b3,
                                     float* __restrict__ out) {
    int lane = threadIdx.x & 31;
    int r = blockIdx.x * 8 + (threadIdx.x >> 5);
    int beg = row_start[r], end = row_start[r + 1];
    float s[NOUT2] = {};
    for (int i = beg + lane; i < end; i += 32) {
        int e = esort[i];
        float w = ew[e];
        const float* p = t12 + (size_t)ecol[e] * NOUT2;
#pragma unroll
        for (int j = 0; j < NOUT2; ++j) s[j] += w * p[j];
    }
#pragma unroll
    for (int j = 0; j < NOUT2; ++j)
        for (int o = 16; o > 0; o >>= 1) s[j] += __shfl_xor(s[j], o, 32);
    if (lane == 0) {
        float x1[NCLS], x2[2];
#pragma unroll
        for (int j = 0; j < NCLS; ++j) x1[j] = s[j] + b2[j];
        x2[0] = s[7] + b3[0];
        x2[1] = s[8] + b3[1];
        float m = x1[0];
#pragma unroll
        for (int j = 1; j < NCLS; ++j) m = fmaxf(m, x1[j]);
        float se = 0.f;
#pragma unroll
        for (int j = 0; j < NCLS; ++j) se += expf(x1[j] - m);
        float lse = m + logf(se);
#pragma unroll
        for (int j = 0; j < NCLS; ++j)
            out[(size_t)r * NCLS + j] = x1[j] - lse;
        float m2   = fmaxf(x2[0], x2[1]);
        float se2  = expf(x2[0] - m2) + expf(x2[1] - m2);
        float lse2 = m2 + logf(se2);
        out[(size_t)N_NODES * NCLS + (size_t)r * 2 + 0] = x2[0] - lse2;
        out[(size_t)N_NODES * NCLS + (size_t)r * 2 + 1] = x2[1] - lse2;
        out[(size_t)N_NODES * NCLS + (size_t)N_NODES * 2 + r] = expf(x1[NCLS - 1] - lse);
    }
}

// ---------------------------------------------------------------------------
extern "C" void kernel_launch(void* const* d_in, const int* in_sizes, int n_in,
                              void* d_out, int out_size, void* d_ws, size_t ws_size,
                              hipStream_t stream) {
    const float* x    = (const float*)d_in[0];
    const int*   erow = (const int*)  d_in[1];
    const int*   ecol = (const int*)  d_in[2];
    const float* ew   = (const float*)d_in[3];
    const float* W1   = (const float*)d_in[4];
    const float* b1   = (const float*)d_in[5];
    const float* W2   = (const float*)d_in[6];
    const float* b2   = (const float*)d_in[7];
    const float* W3   = (const float*)d_in[8];
    const float* b3   = (const float*)d_in[9];
    float* out = (float*)d_out;

    char* ws = (char*)d_ws;
    size_t off = 0;
    auto carve = [&](size_t bytes) -> void* {
        void* p = ws + off;
        off += (bytes + 255) & ~(size_t)255;
        return p;
    };
    float* h0        = (float*)carve(sizeof(float) * (size_t)N_NODES * NHID);
    float* h         = (float*)carve(sizeof(float) * (size_t)N_NODES * NHID);
    float* t12       = (float*)carve(sizeof(float) * (size_t)N_NODES * NOUT2);
    int*   counts    = (int*)  carve(sizeof(int)   * (size_t)N_NODES);
    int*   row_start = (int*)  carve(sizeof(int)   * ((size_t)N_NODES + 1));
    int*   cursor    = (int*)  carve(sizeof(int)   * (size_t)N_NODES);
    int*   esort     = (int*)  carve(sizeof(int)   * (size_t)N_EDGES);
    (void)ws_size; (void)in_sizes; (void)n_in; (void)out_size;

    const int T = 256;
    // CSR build
    k_zero_counts<<<(N_NODES + T - 1) / T, T, 0, stream>>>(counts);
    k_hist<<<(N_EDGES + T - 1) / T, T, 0, stream>>>(erow, counts);
    k_scan<<<1, 1024, 0, stream>>>(counts, row_start);
    k_init_cursor<<<(N_NODES + T - 1) / T, T, 0, stream>>>(row_start, cursor);
    k_scatter<<<(N_EDGES + T - 1) / T, T, 0, stream>>>(erow, cursor, esort);
    // h0 = x @ W1  (WMMA f32, register-tiled 80x32 per wave)
    k_gemm1_wmma<<<dim3(N_NODES / (MT * 16)), 256, 0, stream>>>(x, W1, h0);
    // h = relu(spmm(h0) + b1)
    k_spmm_big<<<N_NODES, NHID, 0, stream>>>(row_start, esort, ecol, ew, h0, b1, h);
    // t12 = h @ [W2|W3]
    k_gemm_small<<<N_NODES / 8, 256, 0, stream>>>(h, W2, W3, t12);
    // heads: spmm + biases + softmaxes
    k_spmm_small_softmax<<<N_NODES / 8, 256, 0, stream>>>(row_start, esort, ecol, ew,
                                                          t12, b2, b3, out);
}